// MaskedChamferLoss_8641474200219
// MI455X (gfx1250) — compile-verified
//
#include <hip/hip_runtime.h>
#include <hip/hip_bf16.h>

typedef float v2f __attribute__((ext_vector_type(2)));
typedef float v8f __attribute__((ext_vector_type(8)));

#define B_  4
#define V1_ 8192            // gt points per batch  (columns of d2)
#define V2_ 8192            // pred points per batch (rows of d2)
#define WAVES_PER_WG 8
#define RTILES 2            // row tiles (16 rows each) per wave
#define ROWS_PER_WG (WAVES_PER_WG * RTILES * 16)   // 256
#define CSPLIT 2            // column-split factor across WGs
#define COLS_PER_WG (V1_ / CSPLIT)                 // 4096
#define NT (COLS_PER_WG / 16)                      // 256 column tiles
#define INF_BITS 0x7F800000u

__global__ void chamfer_init(unsigned* __restrict__ minbits, int n) {
    int i = blockIdx.x * blockDim.x + threadIdx.x;
    if (i < n) minbits[i] = INF_BITS;
}

__global__ __launch_bounds__(256)
void chamfer_main(const float* __restrict__ x_gt,
                  const float* __restrict__ x_pred,
                  unsigned* __restrict__ colmin_bits,   // (B, V1) min over pred
                  unsigned* __restrict__ rowmin_bits)   // (B, V2) min over gt
{
    __shared__ unsigned lds_colmin[COLS_PER_WG];   // 16 KB
    __shared__ unsigned lds_rowmin[ROWS_PER_WG];   // 1 KB

    const int b   = blockIdx.y;
    const int cs  = blockIdx.z;
    const int tid = threadIdx.x;

    for (int i = tid; i < COLS_PER_WG; i += 256) lds_colmin[i] = INF_BITS;
    if (tid < ROWS_PER_WG) lds_rowmin[tid] = INF_BITS;
    __syncthreads();

    const int lane = tid & 31;          // wave32
    const int wave = tid >> 5;
    const int hi   = lane >> 4;         // 16-lane half selects K={0,1} vs K={2,3}
    const int l16  = lane & 15;

    const int wgrowbase = blockIdx.x * ROWS_PER_WG;
    const int colbase   = cs * COLS_PER_WG;

    const float* __restrict__ pred = x_pred + (size_t)b * V2_ * 3;
    const float* __restrict__ gtp  = x_gt   + ((size_t)b * V1_ + colbase) * 3;

    // A (16x4) per row tile: row m = l16, K cols = (-2px,-2py,-2pz, 1)
    // lanes 0-15 hold K={0,1}, lanes 16-31 hold K={2,3}
    // C (16x16) per row tile: C[m][n] = |p_m|^2, VGPR r <-> row r + 8*hi
    v2f amat[RTILES];
    v8f cacc[RTILES];
    float rowmin[RTILES][8];
#pragma unroll
    for (int rt = 0; rt < RTILES; ++rt) {
        const int tbase = wgrowbase + (wave * RTILES + rt) * 16;
        {
            int m = tbase + l16;
            float px = pred[m * 3 + 0];
            float py = pred[m * 3 + 1];
            float pz = pred[m * 3 + 2];
            amat[rt].x = hi ? (-2.0f * pz) : (-2.0f * px);
            amat[rt].y = hi ? 1.0f         : (-2.0f * py);
        }
#pragma unroll
        for (int r = 0; r < 8; ++r) {
            int m = tbase + r + 8 * hi;
            float px = pred[m * 3 + 0];
            float py = pred[m * 3 + 1];
            float pz = pred[m * 3 + 2];
            cacc[rt][r] = px * px + py * py + pz * pz;
            rowmin[rt][r] = __builtin_inff();
        }
    }

    // one 16x16 tile of squared distances: 2 WMMAs + clamp/min reduction
    auto tile_body = [&](int ct, float gx, float gy, float gz) {
        // B (4x16): col n = l16, K rows = (gx, gy, gz, |g|^2)
        float g2 = gx * gx + gy * gy + gz * gz;
        v2f bmat;
        bmat.x = hi ? gz : gx;
        bmat.y = hi ? g2 : gy;

#pragma unroll
        for (int rt = 0; rt < RTILES; ++rt) {
            // D[m][n] = |p_m|^2 - 2 p.g + |g_n|^2
            v8f d = __builtin_amdgcn_wmma_f32_16x16x4_f32(
                false, amat[rt], false, bmat, (short)0, cacc[rt], false, false);

            float v0 = fmaxf(d[0], 0.0f);          // clamp like reference
            float cmin = v0;
            rowmin[rt][0] = fminf(rowmin[rt][0], v0);
#pragma unroll
            for (int r = 1; r < 8; ++r) {
                float v = fmaxf(d[r], 0.0f);
                rowmin[rt][r] = fminf(rowmin[rt][r], v);
                cmin = fminf(cmin, v);
            }
            // all 32 lanes participate: lanes L and L+16 hit the same slot,
            // LDS atomic-min merges the two 8-row halves (no shuffle/branch)
            atomicMin(&lds_colmin[ct * 16 + l16], __float_as_uint(cmin));
        }
    };

    // software-pipelined column sweep: prefetch next gt point while computing
    const float* gp = gtp + l16 * 3;
    float gx = gp[0], gy = gp[1], gz = gp[2];

#pragma unroll 2
    for (int ct = 0; ct < NT - 1; ++ct) {
        gp += 48;                                  // next 16-point tile
        float ngx = gp[0], ngy = gp[1], ngz = gp[2];
        tile_body(ct, gx, gy, gz);
        gx = ngx; gy = ngy; gz = ngz;
    }
    tile_body(NT - 1, gx, gy, gz);                 // peeled last tile

    // merge per-lane row mins (each lane holds min over its 16-column subset)
#pragma unroll
    for (int rt = 0; rt < RTILES; ++rt) {
        const int lbase = (wave * RTILES + rt) * 16;
#pragma unroll
        for (int r = 0; r < 8; ++r)
            atomicMin(&lds_rowmin[lbase + r + 8 * hi],
                      __float_as_uint(rowmin[rt][r]));
    }
    __syncthreads();

    // flush WG-local mins to global (integer min: fully deterministic)
    for (int i = tid; i < COLS_PER_WG; i += 256)
        atomicMin(&colmin_bits[(size_t)b * V1_ + colbase + i], lds_colmin[i]);
    for (int i = tid; i < ROWS_PER_WG; i += 256)
        atomicMin(&rowmin_bits[(size_t)b * V2_ + wgrowbase + i], lds_rowmin[i]);
}

__global__ __launch_bounds__(256)
void chamfer_reduce(const unsigned* __restrict__ rowmin_bits,
                    const unsigned* __restrict__ colmin_bits,
                    const float* __restrict__ mask,
                    float* __restrict__ out)
{
    __shared__ float sA[256];
    __shared__ float sB[256];
    int tid = threadIdx.x;

    float a = 0.0f;   // masked pred->gt sum
    for (int i = tid; i < B_ * V2_; i += 256)
        a += __uint_as_float(rowmin_bits[i]) * mask[i];
    float c = 0.0f;   // gt->pred sum
    for (int i = tid; i < B_ * V1_; i += 256)
        c += __uint_as_float(colmin_bits[i]);

    sA[tid] = a;
    sB[tid] = c;
    __syncthreads();
    for (int s = 128; s > 0; s >>= 1) {
        if (tid < s) { sA[tid] += sA[tid + s]; sB[tid] += sB[tid + s]; }
        __syncthreads();
    }
    if (tid == 0)
        out[0] = sA[0] / (float)(B_ * V2_) + sB[0] / (float)(B_ * V1_);
}

extern "C" void kernel_launch(void* const* d_in, const int* in_sizes, int n_in,
                              void* d_out, int out_size, void* d_ws, size_t ws_size,
                              hipStream_t stream) {
    (void)in_sizes; (void)n_in; (void)out_size; (void)ws_size;
    const float* x_gt   = (const float*)d_in[0];  // (B, V1, 3)
    const float* x_pred = (const float*)d_in[1];  // (B, V2, 3)
    const float* mask   = (const float*)d_in[2];  // (B, V2, 1)

    unsigned* colmin = (unsigned*)d_ws;                                    // B*V1
    unsigned* rowmin = colmin + (size_t)B_ * V1_;                          // B*V2
    float*    out    = (float*)d_out;

    chamfer_init<<<(B_ * (V1_ + V2_) + 255) / 256, 256, 0, stream>>>(
        colmin, B_ * (V1_ + V2_));   // contiguous: colmin then rowmin

    dim3 grid(V2_ / ROWS_PER_WG, B_, CSPLIT);    // (32, 4, 2) = 256 WGs
    chamfer_main<<<grid, 256, 0, stream>>>(x_gt, x_pred, colmin, rowmin);

    chamfer_reduce<<<1, 256, 0, stream>>>(rowmin, colmin, mask, out);
}